// CRFDecoder_58944131170726
// MI455X (gfx1250) — compile-verified
//
#include <hip/hip_runtime.h>
#include <cstdint>

#define BATCH   64
#define SEQ     512
#define IN_DIM  1024
#define NTAGS   50
#define KPAD    64          // padded tag dim (stride of emissions workspace)
#define NEG_INF (-3.0e38f)

#define ROWS_PER_BLOCK 128  // 8 waves x 16 rows
#define KCHUNK         64   // K columns staged per TDM transfer
#define LDS_PITCH      68   // 64 payload + 4 pad dwords (TDM pad) -> bank spread

typedef __attribute__((ext_vector_type(2))) float v2f;
typedef __attribute__((ext_vector_type(8))) float v8f;
typedef __attribute__((ext_vector_type(4))) unsigned int v4u;
typedef __attribute__((ext_vector_type(4))) int          v4i;
typedef __attribute__((ext_vector_type(8))) int          v8i;

// ---------------------------------------------------------------------------
// TDM: 2D tile load (128 rows x 64 f32) from global H into LDS.
// Descriptor per CDNA5 ISA sec 8.3/8.4:
//  g0: count=1 | lds_addr | global_addr[56:0] | type=2
//  g1: data_size=4B, pad_enable (interval 64 dw, amount 4 dw),
//      tensor_dim0=1024, tensor_dim1=32768, tile_dim0=64, tile_dim1=128,
//      tensor_dim0_stride=1024
// ---------------------------------------------------------------------------
__device__ __forceinline__ void tdm_load_tile(unsigned lds_addr, const float* gptr) {
    unsigned long long ga = (unsigned long long)(uintptr_t)gptr;
    v4u g0;
    g0[0] = 1u;                                   // count=1, user descriptor
    g0[1] = lds_addr;                             // LDS byte address
    g0[2] = (unsigned)(ga & 0xFFFFFFFFu);         // global_addr[31:0]
    g0[3] = (unsigned)((ga >> 32) & 0x01FFFFFFu)  // global_addr[56:32]
          | (2u << 30);                           // type=2 ("image")
    v8i g1;
    g1[0] = (int)((2u << 16)                      // data_size: 4 bytes
          | (1u << 20)                            // pad_enable
          | (5u << 22)                            // pad_interval: 64 dwords
          | (3u << 25));                          // pad_amount: 4 dwords
    g1[1] = (int)((unsigned)IN_DIM << 16);        // tensor_dim0[15:0] (=1024)
    g1[2] = (int)((unsigned)(BATCH * SEQ) << 16); // tensor_dim1[15:0] (=32768)
    g1[3] = (int)((unsigned)KCHUNK << 16);        // tile_dim0 = 64
    g1[4] = (int)(unsigned)ROWS_PER_BLOCK;        // tile_dim1 = 128, tile_dim2 = 0
    g1[5] = (int)(unsigned)IN_DIM;                // tensor_dim0_stride[31:0] = 1024
    g1[6] = 0;
    g1[7] = 0;
    v4i z4 = {0, 0, 0, 0};
#if __clang_major__ >= 23
    v8i z8 = {0, 0, 0, 0, 0, 0, 0, 0};
    __builtin_amdgcn_tensor_load_to_lds(g0, g1, z4, z4, z8, 0);
#else
    __builtin_amdgcn_tensor_load_to_lds(g0, g1, z4, z4, 0);
#endif
}

// ---------------------------------------------------------------------------
// Phase 0: pack W [1024,50] into zero-padded K-pair-interleaved Wp2:
//          Wp2[kp*64 + n] = { W[2kp][n], W[2kp+1][n] }, n<50 else 0.
// ---------------------------------------------------------------------------
__global__ void __launch_bounds__(256)
pack_w(const float* __restrict__ W, v2f* __restrict__ Wp2) {
    const int idx = blockIdx.x * 256 + threadIdx.x;   // 0..32767
    const int kp  = idx >> 6;
    const int n   = idx & 63;
    v2f v;
    v.x = (n < NTAGS) ? W[(2 * kp)     * NTAGS + n] : 0.0f;
    v.y = (n < NTAGS) ? W[(2 * kp + 1) * NTAGS + n] : 0.0f;
    Wp2[idx] = v;
}

// ---------------------------------------------------------------------------
// Phase 1: emissions GEMM. TDM double-buffers H tiles (128x64 f32) into LDS;
// each wave owns a 16x64 output tile via V_WMMA_F32_16X16X4_F32.
// Grid: 256 blocks x 256 threads.
// ---------------------------------------------------------------------------
__global__ void __launch_bounds__(256)
emis_gemm_wmma(const float* __restrict__ H,     // [32768, 1024]
               const v2f*   __restrict__ Wp2,   // [512, 64] padded/interleaved
               const float* __restrict__ bias,  // [50]
               float*       __restrict__ E)     // [32768, 64]
{
    __shared__ float Hs[2][ROWS_PER_BLOCK * LDS_PITCH];

    const int lane    = threadIdx.x & 31;
    const int wave    = threadIdx.x >> 5;
    const int l16     = lane & 15;
    const int halfsel = lane >> 4;            // 0: K pair {0,1}, 1: {2,3}
    const int rowblk  = blockIdx.x * ROWS_PER_BLOCK;
    const bool tdm_wave = (threadIdx.x < 32); // wave 0 drives the DMA

    const unsigned lds0 = (unsigned)(uintptr_t)&Hs[0][0];
    const unsigned lds1 = (unsigned)(uintptr_t)&Hs[1][0];
    const int NCH = IN_DIM / KCHUNK;          // 16 chunks

    if (tdm_wave)
        tdm_load_tile(lds0, H + (size_t)rowblk * IN_DIM);

    v8f acc[4] = {v8f{}, v8f{}, v8f{}, v8f{}};

    for (int c = 0; c < NCH; ++c) {
        if (tdm_wave) {
            if (c + 1 < NCH) {
                tdm_load_tile((c & 1) ? lds0 : lds1,
                              H + (size_t)rowblk * IN_DIM + (c + 1) * KCHUNK);
                __builtin_amdgcn_s_wait_tensorcnt(1);   // chunk c landed
            } else {
                __builtin_amdgcn_s_wait_tensorcnt(0);
            }
        }
        __syncthreads();                                // publish buffer c

        const float* hb = &Hs[c & 1][0];
        const int kpbase = (c * KCHUNK) >> 1;
#pragma unroll 4
        for (int kk = 0; kk < KCHUNK; kk += 4) {
            const int ka = kk + halfsel * 2;
            v2f a = *(const v2f*)&hb[(wave * 16 + l16) * LDS_PITCH + ka];
            const int kp = kpbase + (kk >> 1) + halfsel;
#pragma unroll
            for (int nt = 0; nt < 4; ++nt) {
                v2f bm = Wp2[kp * 64 + nt * 16 + l16];
                acc[nt] = __builtin_amdgcn_wmma_f32_16x16x4_f32(
                    false, a, false, bm, (short)0, acc[nt], false, false);
            }
        }
        __syncthreads();       // everyone done reading buffer c before reuse
    }

    // C/D layout: VGPR r -> row (r + 8*halfsel), col = lane%16 (+16*nt)
    const int row0 = rowblk + wave * 16;
#pragma unroll
    for (int nt = 0; nt < 4; ++nt) {
        const int n  = nt * 16 + l16;
        const float bv = (n < NTAGS) ? bias[n] : 0.0f;
#pragma unroll
        for (int r = 0; r < 8; ++r) {
            const int m = r + 8 * halfsel;
            E[(row0 + m) * KPAD + n] = acc[nt][r] + bv;
        }
    }
}

// ---------------------------------------------------------------------------
// Phase 2: gold path score, one thread per batch element (serial over T).
// ---------------------------------------------------------------------------
__global__ void __launch_bounds__(64)
crf_gold(const float* __restrict__ E,
         const int*   __restrict__ tags,
         const uint8_t* __restrict__ mask,
         const float* __restrict__ trans,
         const float* __restrict__ start_t,
         const float* __restrict__ end_t,
         float*       __restrict__ gold)
{
    const int b = threadIdx.x;
    if (b >= BATCH) return;

    int prev = tags[b * SEQ];
    float score = start_t[prev] + E[(b * SEQ) * KPAD + prev];
    for (int t = 1; t < SEQ; ++t) {
        const int   tg = tags[b * SEQ + t];
        const bool  mt = mask[b * SEQ + t] != 0;
        const float st = trans[prev * NTAGS + tg] + E[(b * SEQ + t) * KPAD + tg];
        score += mt ? st : 0.0f;
        prev   = mt ? tg : prev;
    }
    gold[b] = score + end_t[prev];
}

// ---------------------------------------------------------------------------
// Phase 3: forward algorithm + final output. One block (2 waves) per batch,
// thread k owns tag column k; trans[:,k] pinned in VGPRs, alpha via LDS.
// ---------------------------------------------------------------------------
__global__ void __launch_bounds__(64)
crf_forward(const float* __restrict__ E,
            const uint8_t* __restrict__ mask,
            const float* __restrict__ trans,
            const float* __restrict__ start_t,
            const float* __restrict__ end_t,
            const float* __restrict__ gold,
            float*       __restrict__ out)
{
    const int b = blockIdx.x;
    const int k = threadIdx.x;          // 0..63, active < 50

    __shared__ float alpha[KPAD];
    __shared__ float red[KPAD];

    float tc[NTAGS];
    if (k < NTAGS) {
#pragma unroll
        for (int j = 0; j < NTAGS; ++j)
            tc[j] = trans[j * NTAGS + k];
    }

    float ak = NEG_INF;
    if (k < NTAGS)
        ak = start_t[k] + E[(b * SEQ) * KPAD + k];
    alpha[k] = ak;
    __syncthreads();

    for (int t = 1; t < SEQ; ++t) {
        if (k < NTAGS) {
            float m = NEG_INF;
#pragma unroll
            for (int j = 0; j < NTAGS; ++j)
                m = fmaxf(m, alpha[j] + tc[j]);
            float s = 0.0f;
#pragma unroll
            for (int j = 0; j < NTAGS; ++j)
                s += __expf(alpha[j] + tc[j] - m);
            const float nxt = m + __logf(s) + E[(b * SEQ + t) * KPAD + k];
            const bool  mt  = mask[b * SEQ + t] != 0;
            ak = mt ? nxt : ak;
        }
        __syncthreads();
        alpha[k] = ak;
        __syncthreads();
    }

    const float v = (k < NTAGS) ? (ak + end_t[k]) : NEG_INF;
    red[k] = v;
    __syncthreads();
#pragma unroll
    for (int s = 32; s > 0; s >>= 1) {
        if (k < s) red[k] = fmaxf(red[k], red[k + s]);
        __syncthreads();
    }
    const float m = red[0];
    __syncthreads();
    red[k] = __expf(v - m);
    __syncthreads();
#pragma unroll
    for (int s = 32; s > 0; s >>= 1) {
        if (k < s) red[k] += red[k + s];
        __syncthreads();
    }
    if (k == 0)
        out[b] = m + __logf(red[0]) - gold[b];
}

// ---------------------------------------------------------------------------
extern "C" void kernel_launch(void* const* d_in, const int* in_sizes, int n_in,
                              void* d_out, int out_size, void* d_ws, size_t ws_size,
                              hipStream_t stream) {
    const float*   H      = (const float*)  d_in[0];
    const int*     tags   = (const int*)    d_in[1];
    const uint8_t* mask   = (const uint8_t*)d_in[2];
    const float*   W      = (const float*)  d_in[3];
    const float*   bias   = (const float*)  d_in[4];
    const float*   trans  = (const float*)  d_in[5];
    const float*   startt = (const float*)  d_in[6];
    const float*   endt   = (const float*)  d_in[7];
    float*         outp   = (float*)d_out;

    char* ws = (char*)d_ws;
    float* E    = (float*)ws;                                  // 32768*64 f32 = 8 MB
    ws += (size_t)BATCH * SEQ * KPAD * sizeof(float);
    float* gold = (float*)ws;                                  // 64 f32
    ws += 256;                                                 // keep 256B alignment
    v2f*   Wp2  = (v2f*)ws;                                    // 512*64 v2f = 256 KB

    pack_w<<<128, 256, 0, stream>>>(W, Wp2);                   // 32768 elems
    emis_gemm_wmma<<<256, 256, 0, stream>>>(H, Wp2, bias, E);
    crf_gold<<<1, 64, 0, stream>>>(E, tags, mask, trans, startt, endt, gold);
    crf_forward<<<BATCH, 64, 0, stream>>>(E, mask, trans, startt, endt, gold, outp);
}